// GHMCLoss_30751965839586
// MI455X (gfx1250) — compile-verified
//
#include <hip/hip_runtime.h>
#include <hip/hip_bf16.h>

#define BINS 10
#define EPSW 1e-6f
#define TPB 256           // 8 waves of 32
#define MAXBLK 2048

typedef __attribute__((ext_vector_type(2))) float v2f;
typedef __attribute__((ext_vector_type(8))) float v8f;

// ---------------- init: zero the gmax slot ----------------
__global__ void ghm_init_kernel(unsigned* gmax_bits) {
    *gmax_bits = 0u;
}

// ---------------- pass A: gmax = max(|pred - target|) ----------------
__global__ __launch_bounds__(TPB) void ghm_max_kernel(
    const float* __restrict__ pred, const float* __restrict__ targ,
    unsigned* __restrict__ gmax_bits, int N)
{
    const int tid    = blockIdx.x * TPB + threadIdx.x;
    const int stride = gridDim.x * TPB;
    const int n4 = N >> 2;
    const float4* __restrict__ p4 = (const float4*)pred;
    const float4* __restrict__ t4 = (const float4*)targ;

    float m = 0.0f;
    for (int i = tid; i < n4; i += stride) {
        float4 p = p4[i], t = t4[i];
        m = fmaxf(m, fabsf(p.x - t.x));
        m = fmaxf(m, fabsf(p.y - t.y));
        m = fmaxf(m, fabsf(p.z - t.z));
        m = fmaxf(m, fabsf(p.w - t.w));
    }
    for (int i = (n4 << 2) + tid; i < N; i += stride)
        m = fmaxf(m, fabsf(pred[i] - targ[i]));

    // wave32 tree
    for (int off = 16; off > 0; off >>= 1)
        m = fmaxf(m, __shfl_xor(m, off, 32));

    __shared__ float smax[TPB / 32];
    const int lane = threadIdx.x & 31, wave = threadIdx.x >> 5;
    if (lane == 0) smax[wave] = m;
    __syncthreads();
    if (threadIdx.x == 0) {
        float bm = smax[0];
        #pragma unroll
        for (int w = 1; w < TPB / 32; ++w) bm = fmaxf(bm, smax[w]);
        // non-negative floats compare identically as uints -> exact assoc. max
        atomicMax(gmax_bits, __float_as_uint(bm));
    }
}

// ---------------- pass B: sum( w[bin]*d*d ) via WMMA accumulation ----------------
// scale = (BINS-1)/gmax hoisted out of the loop: one v_mul + v_cvt per element
// instead of a 9-op IEEE divide expansion (keeps pass B under the HBM roof).
__device__ __forceinline__ float ghm_val(float d, float scale, const float* wtab) {
    float g = fabsf(d);
    int b = (int)(g * scale);                       // truncate toward zero
    b = b < 0 ? 0 : (b > BINS - 1 ? BINS - 1 : b);  // v_med3_i32
    return wtab[b] * (d * d);
}

__global__ __launch_bounds__(TPB) void ghm_wsum_kernel(
    const float* __restrict__ pred, const float* __restrict__ targ,
    const float* __restrict__ grad_density,
    const unsigned* __restrict__ gmax_bits,
    float* __restrict__ partials, int N)
{
    __shared__ float wtab[BINS];
    __shared__ float swave[TPB / 32];
    if (threadIdx.x < BINS)
        wtab[threadIdx.x] = 1.0f / (grad_density[threadIdx.x] + EPSW);
    __syncthreads();

    const float gmax  = __uint_as_float(*gmax_bits);
    const float scale = (float)(BINS - 1) / gmax;   // one divide per thread
    const int n4     = N >> 2;
    const int tid4   = blockIdx.x * TPB + threadIdx.x;
    const int step4  = gridDim.x * TPB;
    const int nfull4 = (n4 / step4) * step4;   // uniform WMMA region (EXEC all ones)

    const float4* __restrict__ p4 = (const float4*)pred;
    const float4* __restrict__ t4 = (const float4*)targ;

    v8f c = {0.f, 0.f, 0.f, 0.f, 0.f, 0.f, 0.f, 0.f};
    v2f bones; bones[0] = 1.0f; bones[1] = 1.0f;

    for (int i = tid4; i < nfull4; i += step4) {
        float4 p = p4[i], t = t4[i];
        float v0 = ghm_val(p.x - t.x, scale, wtab);
        float v1 = ghm_val(p.y - t.y, scale, wtab);
        float v2 = ghm_val(p.z - t.z, scale, wtab);
        float v3 = ghm_val(p.w - t.w, scale, wtab);
        v2f a; a[0] = v0 + v1; a[1] = v2 + v3;
        // D = A x ones + C : each D[m][n] accumulates rowsum_m; every column
        // of D sums to the wave's running total (16 identical columns).
        c = __builtin_amdgcn_wmma_f32_16x16x4_f32(
                false, a, false, bones, (short)0, c, false, false);
    }

    // non-uniform tails: scalar accumulate (no WMMA -> divergence safe)
    float ts = 0.0f;
    for (int i = nfull4 + tid4; i < n4; i += step4) {
        float4 p = p4[i], t = t4[i];
        ts += ghm_val(p.x - t.x, scale, wtab);
        ts += ghm_val(p.y - t.y, scale, wtab);
        ts += ghm_val(p.z - t.z, scale, wtab);
        ts += ghm_val(p.w - t.w, scale, wtab);
    }
    for (int i = (n4 << 2) + blockIdx.x * TPB + threadIdx.x; i < N; i += gridDim.x * TPB)
        ts += ghm_val(pred[i] - targ[i], scale, wtab);

    // Layout-independent extraction: sum of all 256 D entries == 16 * total.
    float csum = c[0] + c[1] + c[2] + c[3] + c[4] + c[5] + c[6] + c[7];
    float comb = csum * 0.0625f + ts;
    for (int off = 16; off > 0; off >>= 1)
        comb += __shfl_xor(comb, off, 32);

    const int lane = threadIdx.x & 31, wave = threadIdx.x >> 5;
    if (lane == 0) swave[wave] = comb;
    __syncthreads();
    if (threadIdx.x == 0) {
        float s = swave[0];
        #pragma unroll
        for (int w = 1; w < TPB / 32; ++w) s += swave[w];
        partials[blockIdx.x] = s;   // unconditional write: no init needed
    }
}

// ---------------- deterministic final reduction ----------------
__global__ __launch_bounds__(TPB) void ghm_final_kernel(
    const float* __restrict__ partials, int nblocks,
    float* __restrict__ out, float invN)
{
    __shared__ float s[TPB];
    float a = 0.0f;
    for (int i = threadIdx.x; i < nblocks; i += TPB) a += partials[i];
    s[threadIdx.x] = a;
    __syncthreads();
    for (int off = TPB / 2; off > 0; off >>= 1) {
        if (threadIdx.x < off) s[threadIdx.x] += s[threadIdx.x + off];
        __syncthreads();
    }
    if (threadIdx.x == 0) out[0] = s[0] * invN;
}

extern "C" void kernel_launch(void* const* d_in, const int* in_sizes, int n_in,
                              void* d_out, int out_size, void* d_ws, size_t ws_size,
                              hipStream_t stream) {
    const float* pred = (const float*)d_in[0];
    const float* targ = (const float*)d_in[1];
    // d_in[2] = gradient_hist (unused by the reference's returned loss)
    const float* grad_density = (const float*)d_in[3];
    float* out = (float*)d_out;
    const int N = in_sizes[0];

    // workspace layout: [0] gmax bits (u32), [16..] per-block partials
    unsigned* gmax_bits = (unsigned*)d_ws;
    float*    partials  = (float*)d_ws + 16;

    int blocks = MAXBLK;
    int wscap = (int)(ws_size / sizeof(float)) - 16;
    if (wscap < blocks) blocks = wscap > 0 ? wscap : 1;

    ghm_init_kernel<<<1, 1, 0, stream>>>(gmax_bits);
    ghm_max_kernel<<<MAXBLK, TPB, 0, stream>>>(pred, targ, gmax_bits, N);
    ghm_wsum_kernel<<<blocks, TPB, 0, stream>>>(pred, targ, grad_density,
                                                gmax_bits, partials, N);
    ghm_final_kernel<<<1, TPB, 0, stream>>>(partials, blocks, out,
                                            (float)(1.0 / (double)N));
}